// LinODECell_18511309046332
// MI455X (gfx1250) — compile-verified
//
#include <hip/hip_runtime.h>

typedef __attribute__((ext_vector_type(2))) float v2f;
typedef __attribute__((ext_vector_type(8))) float v8f;

#define NTAYLOR 32   // ||dt*A|| <= ~5.7 -> term k=32 ~ 2e-11: fp32-converged

__global__ __launch_bounds__(128) void linode_expm_kernel(
    const float* __restrict__ dt,      // (B,)
    const float* __restrict__ x0,      // (B, 32)
    const float* __restrict__ W,       // (32, 32)
    const float* __restrict__ scalar,  // (1,)
    float* __restrict__ out)           // (B, 32)
{
  const int lane = threadIdx.x & 31;
  const int wave = threadIdx.x >> 5;
  const int b0   = (blockIdx.x * 4 + wave) * 16;  // 16 batch columns per wave
  const int n    = lane & 15;                     // batch column within tile
  const int half = lane >> 4;                     // hi/lo 16-lane group

  const float sc  = scalar[0];
  const float dtv = dt[b0 + n];

  // ---- A (32x32, scaled) in WMMA f32 16x16x4 A-operand layout ----
  // row-tile r (rows r*16..r*16+15), k-chunk c (cols 4c..4c+3):
  //   lane: M = r*16 + n ; v0 = A[M, 4c + 2*half], v1 = A[M, 4c + 2*half + 1]
  v2f a_op[2][8];
#pragma unroll
  for (int r = 0; r < 2; ++r) {
    const int m = r * 16 + n;
#pragma unroll
    for (int c = 0; c < 8; ++c) {
      v2f v = *(const v2f*)(W + m * 32 + 4 * c + 2 * half);
      a_op[r][c] = v * sc;   // kernel = scalar * weight
    }
  }

  // ---- Z = current Taylor term (32x16 tile, B-operand layout) ----
  // chunk c: lane holds Z[4c + 2*half + {0,1}, n] = x0[b0+n, 4c+2*half+{0,1}]
  v2f zb[8], xh[8];
#pragma unroll
  for (int c = 0; c < 8; ++c) {
    v2f v = *(const v2f*)(x0 + (b0 + n) * 32 + 4 * c + 2 * half);
    zb[c] = v;   // t_0 = x0
    xh[c] = v;   // xhat = t_0
  }

#pragma unroll
  for (int it = 1; it <= NTAYLOR; ++it) {
    // Znew = A * Z : two 16x16 output tiles, contracted over 8 k-chunks of 4
    v8f accA = {0.f,0.f,0.f,0.f,0.f,0.f,0.f,0.f};
    v8f accB = {0.f,0.f,0.f,0.f,0.f,0.f,0.f,0.f};
#pragma unroll
    for (int c = 0; c < 8; ++c) {
      accA = __builtin_amdgcn_wmma_f32_16x16x4_f32(
          false, a_op[0][c], false, zb[c], (short)0, accA, false, false);
      accB = __builtin_amdgcn_wmma_f32_16x16x4_f32(
          false, a_op[1][c], false, zb[c], (short)0, accB, false, false);
    }

    // t_k = (dt/k) * A t_{k-1}; 1/it is a compile-time literal after unroll
    const float s = dtv * (1.0f / (float)it);

    // Recycle D-layout -> B-operand layout with half-wave row swaps:
    // D vgpr j = rows {16t+j (lanes0-15), 16t+j+8 (lanes16-31)}
    // after swap(j, j+2): reg j  = {row 16t+j,   row 16t+j+2}   -> B chunk
    //                     reg j+2= {row 16t+j+8, row 16t+j+10}  -> B chunk
    {
      float r0 = accA[0], r1 = accA[1], r2 = accA[2], r3 = accA[3];
      float r4 = accA[4], r5 = accA[5], r6 = accA[6], r7 = accA[7];
      asm volatile("v_permlane16_swap_b32 %0, %1" : "+v"(r0), "+v"(r2));
      asm volatile("v_permlane16_swap_b32 %0, %1" : "+v"(r1), "+v"(r3));
      asm volatile("v_permlane16_swap_b32 %0, %1" : "+v"(r4), "+v"(r6));
      asm volatile("v_permlane16_swap_b32 %0, %1" : "+v"(r5), "+v"(r7));
      v2f c0; c0[0] = r0; c0[1] = r1;   // K = 0/2  (v1: 1/3)
      v2f c1; c1[0] = r4; c1[1] = r5;   // K = 4/6
      v2f c2; c2[0] = r2; c2[1] = r3;   // K = 8/10
      v2f c3; c3[0] = r6; c3[1] = r7;   // K = 12/14
      zb[0] = c0 * s;
      zb[1] = c1 * s;
      zb[2] = c2 * s;
      zb[3] = c3 * s;
    }
    {
      float r0 = accB[0], r1 = accB[1], r2 = accB[2], r3 = accB[3];
      float r4 = accB[4], r5 = accB[5], r6 = accB[6], r7 = accB[7];
      asm volatile("v_permlane16_swap_b32 %0, %1" : "+v"(r0), "+v"(r2));
      asm volatile("v_permlane16_swap_b32 %0, %1" : "+v"(r1), "+v"(r3));
      asm volatile("v_permlane16_swap_b32 %0, %1" : "+v"(r4), "+v"(r6));
      asm volatile("v_permlane16_swap_b32 %0, %1" : "+v"(r5), "+v"(r7));
      v2f c0; c0[0] = r0; c0[1] = r1;   // K = 16/18
      v2f c1; c1[0] = r4; c1[1] = r5;   // K = 20/22
      v2f c2; c2[0] = r2; c2[1] = r3;   // K = 24/26
      v2f c3; c3[0] = r6; c3[1] = r7;   // K = 28/30
      zb[4] = c0 * s;
      zb[5] = c1 * s;
      zb[6] = c2 * s;
      zb[7] = c3 * s;
    }

#pragma unroll
    for (int c = 0; c < 8; ++c) xh[c] += zb[c];  // xhat += t_k
  }

  // xhat store (B-layout -> row-major (B,32)): contiguous 8B per lane
#pragma unroll
  for (int c = 0; c < 8; ++c) {
    *(v2f*)(out + (b0 + n) * 32 + 4 * c + 2 * half) = xh[c];
  }
}

extern "C" void kernel_launch(void* const* d_in, const int* in_sizes, int n_in,
                              void* d_out, int out_size, void* d_ws, size_t ws_size,
                              hipStream_t stream) {
  const float* dt     = (const float*)d_in[0];
  const float* x0     = (const float*)d_in[1];
  const float* weight = (const float*)d_in[2];
  const float* scalar = (const float*)d_in[3];
  float* out = (float*)d_out;

  const int B = in_sizes[0];                 // 65536
  const int blocks = B / (4 * 16);           // 4 waves/block * 16 batch/wave
  linode_expm_kernel<<<blocks, 128, 0, stream>>>(dt, x0, weight, scalar, out);
}